// HGANLayer_19292993093685
// MI455X (gfx1250) — compile-verified
//
#include <hip/hip_runtime.h>
#include <hip/hip_bf16.h>

// ---------------------------------------------------------------------------
// HAN layer for MI455X (gfx1250, wave32).
//  - Dense GEMMs via V_WMMA_F32_16X16X4_F32 with the A tile staged in LDS
//    (shared by the 4 waves of each block; removes 4x redundant global reads
//    and keeps EXEC uniform around the WMMA ops).
//  - Edge softmax: float atomic-max via sign-split int trick, f32 atomic adds.
//  - Message aggregation: one wave per edge, 32-lane coalesced 256-float rows,
//    global_atomic_add_f32 scatter, global_prefetch on the gathered row.
// ---------------------------------------------------------------------------

typedef float v2f __attribute__((ext_vector_type(2)));
typedef float v8f __attribute__((ext_vector_type(8)));

#define NN   50000
#define EE   800000
#define PP   3
#define INF_ 128
#define HH   4
#define DD   64
#define HO   256      // H*OUT
#define HID_ 128
#define KMAX 256      // max K across both GEMMs
#define NEG_SLOPE 0.2f

// ---------------- f32 WMMA GEMM:  C[M,Nc] = A[M,K] * B[K,Nc] ----------------
// Requirements: M %16==0, K %4==0 (K<=KMAX), Nc %64==0.
// block = 128 threads (4 waves), grid = (M/16, Nc/64).
// One 16-row A tile per block staged in LDS; each wave owns one 16x16 C tile.
__global__ void gemm_wmma_f32(const float* __restrict__ A,
                              const float* __restrict__ B,
                              float* __restrict__ C,
                              int M, int K, int Nc) {
    __shared__ float As[16 * KMAX];

    const int lane  = threadIdx.x & 31;
    const int wave  = threadIdx.x >> 5;
    const int mBase = blockIdx.x * 16;
    const int nBase = (blockIdx.y * 4 + wave) * 16;

    // cooperative, coalesced stage of the 16xK A tile (all 128 threads)
    const int tileElems = 16 * K;
    for (int i = threadIdx.x; i < tileElems; i += 128) {
        int r = i / K, c = i - r * K;
        As[i] = A[(size_t)(mBase + r) * K + c];
    }
    __syncthreads();

    const int mlo = lane & 15;                      // A row within tile
    const int n   = nBase + (lane & 15);            // B/C col for this lane
    const int kb  = (lane >> 4) * 2;                // K sub-offset 0 or 2

    v8f acc = {};
    const float* Arow = As + mlo * K;               // LDS row
    for (int k = 0; k < K; k += 4) {
        v2f a, b;
        a.x = Arow[k + kb];                          // ds_load
        a.y = Arow[k + kb + 1];
        b.x = B[(size_t)(k + kb)     * Nc + n];      // global, lane-coalesced
        b.y = B[(size_t)(k + kb + 1) * Nc + n];
        acc = __builtin_amdgcn_wmma_f32_16x16x4_f32(
                  false, a, false, b, (short)0, acc, false, false);
    }
    // C/D layout: VGPR r -> row (r + 8*(lane>=16)), col = lane&15
    const int rowOff = (lane >> 4) * 8;
    #pragma unroll
    for (int r = 0; r < 8; ++r)
        C[(size_t)(mBase + rowOff + r) * Nc + n] = acc[r];
}

// ---------------- helpers ---------------------------------------------------
__global__ void fill_f32(float* __restrict__ p, float v, int count) {
    int i = blockIdx.x * blockDim.x + threadIdx.x;
    if (i < count) p[i] = v;
}

// el[n,h] = feat[n,h,:].attn_l[h,:] ; er likewise
__global__ void el_er_kernel(const float* __restrict__ feat,
                             const float* __restrict__ al,
                             const float* __restrict__ ar,
                             float* __restrict__ el,
                             float* __restrict__ er) {
    int gid = blockIdx.x * blockDim.x + threadIdx.x;
    if (gid >= NN * HH) return;
    int n = gid >> 2, h = gid & 3;
    const float* f = feat + (size_t)n * HO + h * DD;
    const float* L = al + h * DD;
    const float* R = ar + h * DD;
    float sl = 0.f, sr = 0.f;
    for (int d = 0; d < DD; ++d) { float fv = f[d]; sl += fv * L[d]; sr += fv * R[d]; }
    el[gid] = sl;
    er[gid] = sr;
}

__device__ __forceinline__ void atomicMaxF(float* addr, float val) {
    if (val >= 0.0f)
        atomicMax((int*)addr, __float_as_int(val));
    else
        atomicMin((unsigned int*)addr, (unsigned int)__float_as_int(val));
}

__device__ __forceinline__ float edge_e(const float* el, const float* er,
                                        int src, int dst, int h) {
    float e = el[src * HH + h] + er[dst * HH + h];
    return e > 0.f ? e : NEG_SLOPE * e;
}

// pass 1: segment max of leaky-relu'd logits
__global__ void edge_max_kernel(const int* __restrict__ src_idx,
                                const int* __restrict__ dst_idx,
                                const float* __restrict__ el,
                                const float* __restrict__ er,
                                float* __restrict__ mseg) {
    int gid = blockIdx.x * blockDim.x + threadIdx.x;
    if (gid >= EE * HH) return;
    int eid = gid >> 2, h = gid & 3;
    int s = src_idx[eid], d = dst_idx[eid];
    atomicMaxF(&mseg[d * HH + h], edge_e(el, er, s, d, h));
}

__global__ void fix_m_kernel(float* __restrict__ mseg) {
    int gid = blockIdx.x * blockDim.x + threadIdx.x;
    if (gid >= NN * HH) return;
    float v = mseg[gid];
    if (!(v > -3.0e38f)) mseg[gid] = 0.0f;   // empty segment (-inf) -> 0
}

// pass 2: ex = exp(e - m[dst]); segment sum
__global__ void edge_expsum_kernel(const int* __restrict__ src_idx,
                                   const int* __restrict__ dst_idx,
                                   const float* __restrict__ el,
                                   const float* __restrict__ er,
                                   const float* __restrict__ mseg,
                                   float* __restrict__ exbuf,
                                   float* __restrict__ sseg) {
    int gid = blockIdx.x * blockDim.x + threadIdx.x;
    if (gid >= EE * HH) return;
    int eid = gid >> 2, h = gid & 3;
    int s = src_idx[eid], d = dst_idx[eid];
    float ex = __expf(edge_e(el, er, s, d, h) - mseg[d * HH + h]);
    exbuf[gid] = ex;
    atomicAdd(&sseg[d * HH + h], ex);
}

// pass 3: one wave32 per edge; scatter feat[src]*alpha into z[dst]
__global__ void edge_scatter_kernel(const int* __restrict__ src_idx,
                                    const int* __restrict__ dst_idx,
                                    const float* __restrict__ feat,
                                    const float* __restrict__ exbuf,
                                    const float* __restrict__ sseg,
                                    float* __restrict__ zslice) {
    int wave = (blockIdx.x * blockDim.x + threadIdx.x) >> 5;
    int lane = threadIdx.x & 31;
    if (wave >= EE) return;
    int s = src_idx[wave], d = dst_idx[wave];
    const float* frow = feat + (size_t)s * HO;
    float* zrow = zslice + (size_t)d * HO;
    // overlap the random-gather latency with the alpha computation
    __builtin_prefetch(frow + lane * 8, 0, 1);        // global_prefetch_b8
    float alpha[HH];
    #pragma unroll
    for (int h = 0; h < HH; ++h)
        alpha[h] = exbuf[wave * HH + h] / sseg[d * HH + h];
    #pragma unroll
    for (int j = 0; j < HO / 32; ++j) {
        int idx = lane + 32 * j;
        atomicAdd(&zrow[idx], frow[idx] * alpha[idx >> 6]);
    }
}

__global__ void elu_kernel(float* __restrict__ z, int count) {
    int i = blockIdx.x * blockDim.x + threadIdx.x;
    if (i < count) { float x = z[i]; z[i] = x > 0.f ? x : expm1f(x); }
}

// w_sem[r] = tanh(t[r,:] + b1) . w2      (r = p*N + n)
__global__ void tanh_dot_kernel(const float* __restrict__ t,
                                const float* __restrict__ b1,
                                const float* __restrict__ w2,
                                float* __restrict__ w_sem) {
    int r = blockIdx.x * blockDim.x + threadIdx.x;
    if (r >= PP * NN) return;
    const float* row = t + (size_t)r * HID_;
    float acc = 0.f;
    for (int k = 0; k < HID_; ++k)
        acc += tanhf(row[k] + b1[k]) * w2[k];
    w_sem[r] = acc;
}

// softmax over P and weighted combine: out[n,d] = sum_p beta[n,p]*z[p,n,d]
__global__ void combine_kernel(const float* __restrict__ z_all,
                               const float* __restrict__ w_sem,
                               float* __restrict__ out) {
    int gid = blockIdx.x * blockDim.x + threadIdx.x;
    if (gid >= NN * HO) return;
    int n = gid >> 8, d = gid & 255;
    float w0 = w_sem[0 * NN + n];
    float w1 = w_sem[1 * NN + n];
    float w2 = w_sem[2 * NN + n];
    float mx = fmaxf(w0, fmaxf(w1, w2));
    float e0 = __expf(w0 - mx), e1 = __expf(w1 - mx), e2 = __expf(w2 - mx);
    float inv = 1.0f / (e0 + e1 + e2);
    size_t off = (size_t)n * HO + d;
    float v = e0 * inv * z_all[off]
            + e1 * inv * z_all[(size_t)NN * HO + off]
            + e2 * inv * z_all[2 * (size_t)NN * HO + off];
    out[gid] = v;
}

// ---------------------------------------------------------------------------
extern "C" void kernel_launch(void* const* d_in, const int* in_sizes, int n_in,
                              void* d_out, int out_size, void* d_ws, size_t ws_size,
                              hipStream_t stream) {
    const float* h      = (const float*)d_in[0];            // [N,128]
    const int*   edges  = (const int*)  d_in[1];            // [P,2,E]
    const float* fc_w   = (const float*)d_in[2];            // [P,128,256]
    const float* attn_l = (const float*)d_in[3];            // [P,4,64]
    const float* attn_r = (const float*)d_in[4];            // [P,4,64]
    const float* w1     = (const float*)d_in[5];            // [256,128]
    const float* b1     = (const float*)d_in[6];            // [128]
    const float* w2     = (const float*)d_in[7];            // [128]
    float* out = (float*)d_out;                             // [N,256]

    // workspace layout (floats)
    float* ws     = (float*)d_ws;
    float* z_all  = ws;                                     // P*N*256
    float* scratch= z_all + (size_t)PP * NN * HO;           // phase-A region
    float* featp  = scratch;                                // N*256
    float* el     = featp + (size_t)NN * HO;                // N*4
    float* er     = el + NN * HH;                           // N*4
    float* mseg   = er + NN * HH;                           // N*4
    float* sseg   = mseg + NN * HH;                         // N*4
    float* exbuf  = sseg + NN * HH;                         // E*4
    float* t_sem  = scratch;                                // P*N*128 (reuses phase-A)
    float* w_sem  = scratch + (size_t)PP * NN * HID_;       // P*N

    const int blk = 256;

    for (int p = 0; p < PP; ++p) {
        const int* src_idx = edges + (size_t)p * 2 * EE;
        const int* dst_idx = src_idx + EE;
        float* zslice = z_all + (size_t)p * NN * HO;

        // feat = h @ fc_w[p]   (WMMA f32, LDS-staged A tile)
        gemm_wmma_f32<<<dim3(NN / 16, HO / 64), 128, 0, stream>>>(
            h, fc_w + (size_t)p * INF_ * HO, featp, NN, INF_, HO);

        // el / er
        el_er_kernel<<<(NN * HH + blk - 1) / blk, blk, 0, stream>>>(
            featp, attn_l + (size_t)p * HH * DD, attn_r + (size_t)p * HH * DD, el, er);

        // init segment buffers + z slice
        fill_f32<<<(NN * HH + blk - 1) / blk, blk, 0, stream>>>(mseg, -__builtin_huge_valf(), NN * HH);
        fill_f32<<<(NN * HH + blk - 1) / blk, blk, 0, stream>>>(sseg, 0.0f, NN * HH);
        fill_f32<<<(NN * HO + blk - 1) / blk, blk, 0, stream>>>(zslice, 0.0f, NN * HO);

        // edge softmax
        edge_max_kernel<<<(EE * HH + blk - 1) / blk, blk, 0, stream>>>(
            src_idx, dst_idx, el, er, mseg);
        fix_m_kernel<<<(NN * HH + blk - 1) / blk, blk, 0, stream>>>(mseg);
        edge_expsum_kernel<<<(EE * HH + blk - 1) / blk, blk, 0, stream>>>(
            src_idx, dst_idx, el, er, mseg, exbuf, sseg);

        // message aggregation (wave per edge)
        edge_scatter_kernel<<<(EE * 32 + blk - 1) / blk, blk, 0, stream>>>(
            src_idx, dst_idx, featp, exbuf, sseg, zslice);

        // ELU
        elu_kernel<<<(NN * HO + blk - 1) / blk, blk, 0, stream>>>(zslice, NN * HO);
    }

    // semantic attention: t = z @ w1  (WMMA f32, M = P*N rows)
    gemm_wmma_f32<<<dim3((PP * NN) / 16, HID_ / 64), 128, 0, stream>>>(
        z_all, w1, t_sem, PP * NN, HO, HID_);

    tanh_dot_kernel<<<(PP * NN + blk - 1) / blk, blk, 0, stream>>>(t_sem, b1, w2, w_sem);

    combine_kernel<<<(NN * HO + blk - 1) / blk, blk, 0, stream>>>(z_all, w_sem, out);
}